// GNNModel_33449205301450
// MI455X (gfx1250) — compile-verified
//
#include <hip/hip_runtime.h>

// GCN forward for MI455X (gfx1250, wave32).
//   layer: h = X@W (WMMA fp32) ; agg[dst] += norm * h[src] (fp32 atomics, L2-resident)
//          out = relu(agg + h*dinv^2 + b)
//   then mean-pool over nodes and a 128x64 FC.
//
// ws layout (floats): bufA[N*128] | bufB[N*128] | dinv[N] | pooled[128]
// ~103 MB total; both 51.2 MB feature buffers fit in the 192 MB L2.

typedef __attribute__((ext_vector_type(2))) float v2f;
typedef __attribute__((ext_vector_type(8))) float v8f;

#define IN_CH  128
#define OUT_FC 64

__device__ __forceinline__ void atomicAddF32(float* addr, float val) {
  // no-return fp32 atomic add (STOREcnt-tracked); avoids any CAS fallback
  asm volatile("global_atomic_add_f32 %0, %1, off" : : "v"(addr), "v"(val) : "memory");
}

// ---------------- degree ----------------
__global__ __launch_bounds__(256) void deg_kernel(const int* __restrict__ dst,
                                                  float* __restrict__ deg, int E) {
  int e = blockIdx.x * blockDim.x + threadIdx.x;
  if (e < E) atomicAddF32(&deg[dst[e]], 1.0f);
}

__global__ __launch_bounds__(256) void dinv_kernel(float* __restrict__ deg, int N) {
  int i = blockIdx.x * blockDim.x + threadIdx.x;
  if (i < N) deg[i] = rsqrtf(deg[i] + 1.0f);   // in place: deg -> dinv
}

// ---------------- GEMM: C[M,128] = A[M,128] @ B[128,128], fp32 WMMA ----------------
// One wave per 16x16 tile. Block = 256 threads = 8 waves = one 16-row strip of all
// 8 column tiles (A rows reused 8x out of L2/L0).
__global__ __launch_bounds__(256) void gemm_wmma_f32(const float* __restrict__ A,
                                                     const float* __restrict__ B,
                                                     float* __restrict__ C) {
  const int lane = threadIdx.x & 31;
  const int nBase = (threadIdx.x >> 5) * 16;     // wave id -> N tile
  const int mBase = blockIdx.x * 16;
  const int lo = lane & 15;
  const int hi = lane >> 4;

  // A fragment: lane holds row (mBase+lo), K = k0 + 2*hi + {0,1}  -> contiguous b64 load
  const float* aRow = A + (size_t)(mBase + lo) * IN_CH + 2 * hi;
  // B fragment: lane holds col (nBase+lo), rows K = k0 + 2*hi + {0,1}
  const float* bCol = B + nBase + lo;

  v8f acc = {};
#pragma unroll
  for (int k0 = 0; k0 < IN_CH; k0 += 4) {
    v2f a = *(const v2f*)(aRow + k0);
    v2f b;
    b.x = bCol[(size_t)(k0 + 2 * hi) * IN_CH];
    b.y = bCol[(size_t)(k0 + 2 * hi + 1) * IN_CH];
    acc = __builtin_amdgcn_wmma_f32_16x16x4_f32(false, a, false, b,
                                                (short)0, acc, false, false);
  }

  // C tile: vgpr r -> row mBase + r + 8*hi, col nBase + lo (stores coalesce over lanes)
  float* cBase = C + (size_t)(mBase + 8 * hi) * IN_CH + nBase + lo;
#pragma unroll
  for (int r = 0; r < 8; ++r) cBase[(size_t)r * IN_CH] = acc[r];
}

// ---------------- edge aggregation: one wave per edge ----------------
// lane l owns channels 4l..4l+3: float4 gather of h[src], 4 fp32 atomic adds to agg[dst]
__global__ __launch_bounds__(256) void agg_kernel(const float* __restrict__ H,
                                                  const int* __restrict__ src,
                                                  const int* __restrict__ dst,
                                                  const float* __restrict__ dinv,
                                                  float* __restrict__ agg, int E) {
  int e = (int)((blockIdx.x * (unsigned)blockDim.x + threadIdx.x) >> 5);
  int lane = threadIdx.x & 31;
  if (e >= E) return;
  int s = src[e];
  int d = dst[e];
  float norm = dinv[s] * dinv[d];
  float4 hv = *((const float4*)(H + (size_t)s * IN_CH) + lane);
  float* ap = agg + (size_t)d * IN_CH + 4 * lane;
  atomicAddF32(ap + 0, hv.x * norm);
  atomicAddF32(ap + 1, hv.y * norm);
  atomicAddF32(ap + 2, hv.z * norm);
  atomicAddF32(ap + 3, hv.w * norm);
}

// ---------------- post: agg = relu(agg + h*dinv^2 + bias), in place ----------------
__global__ __launch_bounds__(256) void post_kernel(float* __restrict__ agg,
                                                   const float* __restrict__ h,
                                                   const float* __restrict__ dinv,
                                                   const float* __restrict__ bias,
                                                   int N) {
  int t = blockIdx.x * blockDim.x + threadIdx.x;     // over N*32 float4's
  if (t >= N * 32) return;
  int row = t >> 5;
  int q = t & 31;
  float di = dinv[row];
  float self = di * di;
  float4 a = ((const float4*)agg)[t];
  float4 hv = ((const float4*)h)[t];
  float4 bv = ((const float4*)bias)[q];
  float4 r;
  r.x = fmaxf(fmaf(hv.x, self, a.x) + bv.x, 0.0f);
  r.y = fmaxf(fmaf(hv.y, self, a.y) + bv.y, 0.0f);
  r.z = fmaxf(fmaf(hv.z, self, a.z) + bv.z, 0.0f);
  r.w = fmaxf(fmaf(hv.w, self, a.w) + bv.w, 0.0f);
  ((float4*)agg)[t] = r;
}

// ---------------- column-sum pool ----------------
__global__ __launch_bounds__(128) void pool_kernel(const float* __restrict__ h,
                                                   float* __restrict__ pooled, int N) {
  int c = threadIdx.x;                        // channel 0..127
  float s = 0.0f;
  for (int r = blockIdx.x; r < N; r += gridDim.x) s += h[(size_t)r * IN_CH + c];
  atomicAddF32(&pooled[c], s);
}

// ---------------- FC head: out[64] = (pooled/N) @ Wfc + bfc ----------------
__global__ __launch_bounds__(64) void fc_kernel(const float* __restrict__ pooled,
                                                const float* __restrict__ Wfc,
                                                const float* __restrict__ bfc,
                                                float* __restrict__ out, float invN) {
  int j = threadIdx.x;
  if (j >= OUT_FC) return;
  float s = 0.0f;
#pragma unroll 4
  for (int c = 0; c < IN_CH; ++c) s += pooled[c] * Wfc[c * OUT_FC + j];
  out[j] = s * invN + bfc[j];
}

extern "C" void kernel_launch(void* const* d_in, const int* in_sizes, int n_in,
                              void* d_out, int out_size, void* d_ws, size_t ws_size,
                              hipStream_t stream) {
  const float* x   = (const float*)d_in[0];
  const int*   ei  = (const int*)d_in[1];   // [2,E] row-major: src then dst
  const float* W1  = (const float*)d_in[2];
  const float* b1  = (const float*)d_in[3];
  const float* W2  = (const float*)d_in[4];
  const float* b2  = (const float*)d_in[5];
  const float* Wfc = (const float*)d_in[6];
  const float* bfc = (const float*)d_in[7];

  const int N = in_sizes[0] / IN_CH;        // 100000 (multiple of 16)
  const int E = in_sizes[1] / 2;            // 1600000
  const int* srcIdx = ei;
  const int* dstIdx = ei + E;

  float* bufA   = (float*)d_ws;             // GEMM outputs (h1, then g2)
  float* bufB   = bufA + (size_t)N * IN_CH; // agg1 -> h(relu) -> agg2 -> h2
  float* dinv   = bufB + (size_t)N * IN_CH; // deg -> dinv (in place)
  float* pooled = dinv + N;

  const size_t featBytes = (size_t)N * IN_CH * sizeof(float);
  const int THREADS = 256;
  const int mTiles = N / 16;                       // 6250
  const int edgeWaveBlocks = (E * 32 + THREADS - 1) / THREADS;
  const int postBlocks = (N * 32 + THREADS - 1) / THREADS;

  // ---- degree / dinv ----
  hipMemsetAsync(dinv, 0, (size_t)N * sizeof(float), stream);
  deg_kernel<<<(E + THREADS - 1) / THREADS, THREADS, 0, stream>>>(dstIdx, dinv, E);
  dinv_kernel<<<(N + THREADS - 1) / THREADS, THREADS, 0, stream>>>(dinv, N);

  // ---- layer 1 ----
  gemm_wmma_f32<<<mTiles, THREADS, 0, stream>>>(x, W1, bufA);            // h1 = x@W1
  hipMemsetAsync(bufB, 0, featBytes, stream);
  agg_kernel<<<edgeWaveBlocks, THREADS, 0, stream>>>(bufA, srcIdx, dstIdx, dinv, bufB, E);
  post_kernel<<<postBlocks, THREADS, 0, stream>>>(bufB, bufA, dinv, b1, N); // bufB = relu(...)

  // ---- layer 2 ----
  gemm_wmma_f32<<<mTiles, THREADS, 0, stream>>>(bufB, W2, bufA);         // g2 = h@W2
  hipMemsetAsync(bufB, 0, featBytes, stream);
  agg_kernel<<<edgeWaveBlocks, THREADS, 0, stream>>>(bufA, srcIdx, dstIdx, dinv, bufB, E);
  post_kernel<<<postBlocks, THREADS, 0, stream>>>(bufB, bufA, dinv, b2, N); // bufB = h2

  // ---- pool + FC ----
  hipMemsetAsync(pooled, 0, IN_CH * sizeof(float), stream);
  pool_kernel<<<512, 128, 0, stream>>>(bufB, pooled, N);
  fc_kernel<<<1, 64, 0, stream>>>(pooled, Wfc, bfc, (float*)d_out, 1.0f / (float)N);
}